// VariationalDeepGraphClustering_33457795236116
// MI455X (gfx1250) — compile-verified
//
#include <hip/hip_runtime.h>
#include <hip/hip_bf16.h>

// ---------------------------------------------------------------------------
// VariationalDeepGraphClustering forward for MI455X (gfx1250, wave32, WMMA)
//
// N=100000, E=1600000, D_IN=256, H1=128, H2=64, ZDIM=32, K=50.
//
// Dense GEMMs: V_WMMA_F32_16X16X4_F32, 16x32 C super-tile per wave32
// (one A fetch feeds two WMMAs -> 2x A reuse). Specialized via templates
// so ReLU/bias fusion is branch-free in the inner loop.
// Graph aggregation: float4 gathers + L2-resident f32 atomics.
// ---------------------------------------------------------------------------

typedef __attribute__((ext_vector_type(2))) float v2f;
typedef __attribute__((ext_vector_type(8))) float v8f;

// ---------------- generic fill ----------------
__global__ void k_fill(float* __restrict__ p, float v, long long n) {
  long long i = (long long)blockIdx.x * blockDim.x + threadIdx.x;
  if (i < n) p[i] = v;
}

// ---------------- degree (self-loop weight pre-filled as 1.0) ----------------
__global__ void k_deg(const int* __restrict__ row, const float* __restrict__ w,
                      float* __restrict__ deg, int e) {
  int i = blockIdx.x * blockDim.x + threadIdx.x;
  if (i < e) atomicAdd(&deg[row[i]], w[i]);
}

// deg -> dinv (in place), selfnorm = dinv^2
__global__ void k_dinv(float* __restrict__ dinv, float* __restrict__ selfn, int n) {
  int i = blockIdx.x * blockDim.x + threadIdx.x;
  if (i < n) {
    float d = dinv[i];
    float dv = (d > 0.f) ? rsqrtf(fmaxf(d, 1e-12f)) : 0.f;
    dinv[i] = dv;
    selfn[i] = dv * dv;
  }
}

// per-edge symmetric GCN norm
__global__ void k_norm(const int* __restrict__ row, const int* __restrict__ col,
                       const float* __restrict__ w, const float* __restrict__ dinv,
                       float* __restrict__ nrm, int e) {
  int i = blockIdx.x * blockDim.x + threadIdx.x;
  if (i < e) nrm[i] = dinv[row[i]] * w[i] * dinv[col[i]];
}

// ---------------- f32 WMMA GEMM ----------------
// C[M,Nc] = op(A[M,K]) @ B[K,Nc] (+bias, +relu), one 16x32 tile per wave32.
// V_WMMA_F32_16X16X4_F32 operand layout (ISA 7.12.2):
//   A 16x4 : lanes 0-15 row M=lane  K={k,k+1}; lanes 16-31 row M=lane-16 K={k+2,k+3}
//   B 4x16 : mirrored (lane&15 = column, lane>>4 selects the K pair)
//   C      : VGPR r -> row tm + (lane>>4)*8 + r, col tn + (lane&15)
// Requires: M%16==0, Nc%32==0, K%4==0 (true for every layer here).
template <int RELU_A, int RELU_OUT, int BIAS>
__global__ void gemm_f32_wmma(const float* __restrict__ A, const float* __restrict__ B,
                              const float* __restrict__ bias, float* __restrict__ C,
                              int M, int K, int Nc) {
  const int wave = blockIdx.x * (blockDim.x >> 5) + (threadIdx.x >> 5);
  const int lane = threadIdx.x & 31;
  const int tilesN = Nc >> 5;                    // 32-wide super-tiles
  const int tilesTotal = (M >> 4) * tilesN;
  if (wave >= tilesTotal) return;                // uniform per wave: EXEC all-1s
  const int tm = (wave / tilesN) << 4;
  const int tn = (wave % tilesN) << 5;

  const int half = lane >> 4;                    // selects K pair {k,k+1} vs {k+2,k+3}
  const int l15  = lane & 15;
  const int koff = half << 1;

  const float* __restrict__ pA = A + (size_t)(tm + l15) * K + koff;
  const float* __restrict__ pB = B + (size_t)koff * Nc + tn + l15;

  v8f acc0 = {}, acc1 = {};
  for (int k = 0; k < K; k += 4) {
    float a0 = pA[0];
    float a1 = pA[1];
    if (RELU_A) { a0 = fmaxf(a0, 0.f); a1 = fmaxf(a1, 0.f); }
    v2f a; a.x = a0; a.y = a1;
    v2f b0, b1;
    b0.x = pB[0];   b0.y = pB[Nc];               // columns tn..tn+15
    b1.x = pB[16];  b1.y = pB[Nc + 16];          // columns tn+16..tn+31
    // (neg_a, A, neg_b, B, c_mod, C, reuse_a, reuse_b)
    acc0 = __builtin_amdgcn_wmma_f32_16x16x4_f32(false, a, false, b0, (short)0, acc0,
                                                 false, false);
    acc1 = __builtin_amdgcn_wmma_f32_16x16x4_f32(false, a, false, b1, (short)0, acc1,
                                                 false, false);
    pA += 4;
    pB += (size_t)Nc * 4;
  }

  const float bc0 = BIAS ? bias[tn + l15] : 0.f;
  const float bc1 = BIAS ? bias[tn + l15 + 16] : 0.f;
  float* pC = C + (size_t)(tm + (half << 3)) * Nc + tn + l15;
  for (int r = 0; r < 8; ++r) {
    float v0 = acc0[r] + bc0;
    float v1 = acc1[r] + bc1;
    if (RELU_OUT) { v0 = fmaxf(v0, 0.f); v1 = fmaxf(v1, 0.f); }
    pC[0]  = v0;
    pC[16] = v1;
    pC += Nc;
  }
}

// ---------------- GCN aggregation ----------------
// init: agg = t * selfnorm[node] + bias   (self-loop term + bias), float4-wide
__global__ void k_agg_init(const float4* __restrict__ t4, const float* __restrict__ selfn,
                           const float* __restrict__ bias, float4* __restrict__ agg4,
                           int ldq, long long total4) {
  long long i = (long long)blockIdx.x * blockDim.x + threadIdx.x;
  if (i < total4) {
    long long node = i >> ldq;
    int f4 = (int)(i & ((1 << ldq) - 1)) << 2;
    float s = selfn[node];
    float4 t = t4[i];
    float4 o;
    o.x = t.x * s + bias[f4 + 0];
    o.y = t.y * s + bias[f4 + 1];
    o.z = t.z * s + bias[f4 + 2];
    o.w = t.w * s + bias[f4 + 3];
    agg4[i] = o;
  }
}

// edges: agg[row] += t[col] * norm ; 4 features per thread (b128 gather +
// 4 f32 atomics, all L2-resident).
__global__ void k_agg_edges(const int* __restrict__ row, const int* __restrict__ col,
                            const float* __restrict__ nrm, const float4* __restrict__ t4,
                            float* __restrict__ agg, int ld, long long total4) {
  long long i = (long long)blockIdx.x * blockDim.x + threadIdx.x;
  if (i < total4) {
    const int ldq = ld - 2;
    int q4 = (int)(i & ((1 << ldq) - 1));        // which float4 within the feature dim
    int e  = (int)(i >> ldq);
    float w = nrm[e];
    float4 t = t4[(((long long)col[e]) << ldq) + q4];
    float* dst = agg + (((long long)row[e]) << ld) + (q4 << 2);
    atomicAdd(dst + 0, t.x * w);
    atomicAdd(dst + 1, t.y * w);
    atomicAdd(dst + 2, t.z * w);
    atomicAdd(dst + 3, t.w * w);
  }
}

// ---------------- BatchNorm (training-mode batch stats, biased variance) ----
__global__ void k_bn_stats(const float* __restrict__ x, float* __restrict__ sums,
                           int n, int d) {
  int tid = blockIdx.x * blockDim.x + threadIdx.x;
  int nth = gridDim.x * blockDim.x;              // launched so nth % d == 0
  int c = tid % d;
  int r0 = tid / d;
  int step = nth / d;
  float s = 0.f, s2 = 0.f;
  for (int r = r0; r < n; r += step) {
    float v = x[(long long)r * d + c];
    s += v; s2 += v * v;
  }
  atomicAdd(&sums[c], s);
  atomicAdd(&sums[d + c], s2);
}

__global__ void k_bn_coef(const float* __restrict__ sums, const float* __restrict__ g,
                          const float* __restrict__ beta, float* __restrict__ coef,
                          int n, int d) {
  int c = blockIdx.x * blockDim.x + threadIdx.x;
  if (c < d) {
    float fn = (float)n;
    float mean = sums[c] / fn;
    float var = sums[d + c] / fn - mean * mean;
    float sc = g[c] * rsqrtf(var + 1e-5f);
    coef[c] = sc;
    coef[d + c] = beta[c] - mean * sc;
  }
}

// y = x*scale + shift; optional second destination (z == mu). In-place legal.
__global__ void k_bn_apply(const float* x, const float* coef,
                           float* y0, float* y1, int ld, long long total) {
  long long i = (long long)blockIdx.x * blockDim.x + threadIdx.x;
  if (i < total) {
    int c = (int)(i & ((1 << ld) - 1));
    float v = x[i] * coef[c] + coef[(1 << ld) + c];
    y0[i] = v;
    if (y1) y1[i] = v;
  }
}

// ---------------- Student-t cluster assignment ----------------
__global__ void k_q(const float* __restrict__ z, const float* __restrict__ centers,
                    float* __restrict__ q, int n) {
  __shared__ float cs[50 * 32];
  __shared__ float cn[50];
  for (int i = threadIdx.x; i < 50 * 32; i += blockDim.x) cs[i] = centers[i];
  __syncthreads();
  for (int k = threadIdx.x; k < 50; k += blockDim.x) {
    float s = 0.f;
    for (int j = 0; j < 32; ++j) { float c = cs[k * 32 + j]; s += c * c; }
    cn[k] = s;
  }
  __syncthreads();
  int i = blockIdx.x * blockDim.x + threadIdx.x;
  if (i >= n) return;
  float zi[32]; float zn = 0.f;
  for (int j = 0; j < 32; ++j) { float v = z[(long long)i * 32 + j]; zi[j] = v; zn += v * v; }
  float ssum = 0.f;
  for (int k = 0; k < 50; ++k) {
    float dot = 0.f;
    for (int j = 0; j < 32; ++j) dot += zi[j] * cs[k * 32 + j];
    float dd = fmaxf(zn + cn[k] - 2.f * dot, 0.f);
    ssum += 1.f / (1.f + dd);
  }
  float inv = 1.f / ssum;
  for (int k = 0; k < 50; ++k) {
    float dot = 0.f;
    for (int j = 0; j < 32; ++j) dot += zi[j] * cs[k * 32 + j];
    float dd = fmaxf(zn + cn[k] - 2.f * dot, 0.f);
    q[(long long)i * 50 + k] = (1.f / (1.f + dd)) * inv;
  }
}

// ---------------------------------------------------------------------------
extern "C" void kernel_launch(void* const* d_in, const int* in_sizes, int n_in,
                              void* d_out, int out_size, void* d_ws, size_t ws_size,
                              hipStream_t stream) {
  constexpr int N_ = 100000, E_ = 1600000;
  (void)in_sizes; (void)n_in; (void)out_size; (void)ws_size;

  const float* x       = (const float*)d_in[0];
  const int*   eidx    = (const int*)  d_in[1];
  const float* ew      = (const float*)d_in[2];
  const float* W_g1    = (const float*)d_in[3];
  const float* b_g1    = (const float*)d_in[4];
  const float* W_g2    = (const float*)d_in[5];
  const float* b_g2    = (const float*)d_in[6];
  const float* W_g3    = (const float*)d_in[7];
  const float* b_g3    = (const float*)d_in[8];
  const float* W_mu    = (const float*)d_in[9];
  const float* b_mu    = (const float*)d_in[10];
  const float* W_lv    = (const float*)d_in[11];
  const float* b_lv    = (const float*)d_in[12];
  const float* g_mu    = (const float*)d_in[13];
  const float* beta_mu = (const float*)d_in[14];
  const float* g_lv    = (const float*)d_in[15];
  const float* beta_lv = (const float*)d_in[16];
  const float* W_d1    = (const float*)d_in[17];
  const float* b_d1    = (const float*)d_in[18];
  const float* g_d1    = (const float*)d_in[19];
  const float* beta_d1 = (const float*)d_in[20];
  const float* W_d2    = (const float*)d_in[21];
  const float* b_d2    = (const float*)d_in[22];
  const float* g_d2    = (const float*)d_in[23];
  const float* beta_d2 = (const float*)d_in[24];
  const float* W_d3    = (const float*)d_in[25];
  const float* b_d3    = (const float*)d_in[26];
  const float* centers = (const float*)d_in[27];

  const int* row = eidx;        // edge_index[0]
  const int* col = eidx + E_;   // edge_index[1]

  // output layout: z | mu | logvar | x_recon | q
  float* out    = (float*)d_out;
  float* out_z  = out;
  float* out_mu = out + (long long)N_ * 32;
  float* out_lv = out + 2LL * N_ * 32;
  float* out_xr = out + 3LL * N_ * 32;
  float* out_q  = out + 3LL * N_ * 32 + (long long)N_ * 256;

  // workspace carve-up (~110 MB of f32); all pieces 16B aligned
  float* ws    = (float*)d_ws;
  float* P0    = ws;                          // N*128 arena A
  float* P1    = P0 + (long long)N_ * 128;    // N*128 arena B
  float* nrm   = P1 + (long long)N_ * 128;    // E
  float* dinv  = nrm + E_;                    // N (deg, then dinv in place)
  float* selfn = dinv + N_;                   // N
  float* stats = selfn + N_;                  // 2*128 max (sum, sumsq)
  float* coef  = stats + 512;                 // 2*128 max (scale, shift)

  auto cdiv = [](long long a, long long b) { return (unsigned)((a + b - 1) / b); };

  // --- GCN normalization ---
  k_fill<<<cdiv(N_, 256), 256, 0, stream>>>(dinv, 1.0f, N_);  // self-loop weight
  k_deg <<<cdiv(E_, 256), 256, 0, stream>>>(row, ew, dinv, E_);
  k_dinv<<<cdiv(N_, 256), 256, 0, stream>>>(dinv, selfn, N_);
  k_norm<<<cdiv(E_, 256), 256, 0, stream>>>(row, col, ew, dinv, nrm, E_);

  auto gemm_blocks = [&](int M, int Nc) { return cdiv((long long)(M >> 4) * (Nc >> 5), 4); };
  auto bn = [&](const float* xin, const float* g, const float* beta,
                float* y0, float* y1, int ld) {
    int d = 1 << ld;
    k_fill    <<<1, 256, 0, stream>>>(stats, 0.f, 2 * d);
    k_bn_stats<<<512, 256, 0, stream>>>(xin, stats, N_, d);  // 131072 threads, %d==0
    k_bn_coef <<<1, 256, 0, stream>>>(stats, g, beta, coef, N_, d);
    k_bn_apply<<<cdiv((long long)N_ * d, 256), 256, 0, stream>>>(
        xin, coef, y0, y1, ld, (long long)N_ * d);
  };
  auto gcn_agg = [&](const float* t, const float* bias, float* agg, int ld) {
    long long total4 = (long long)N_ << (ld - 2);
    k_agg_init<<<cdiv(total4, 256), 256, 0, stream>>>(
        (const float4*)t, selfn, bias, (float4*)agg, ld - 2, total4);
    long long etotal4 = (long long)E_ << (ld - 2);
    k_agg_edges<<<cdiv(etotal4, 256), 256, 0, stream>>>(
        row, col, nrm, (const float4*)t, agg, ld, etotal4);
  };

  // --- GCN encoder (ReLU fused into next GEMM's A-load) ---
  gemm_f32_wmma<0, 0, 0><<<gemm_blocks(N_, 128), 128, 0, stream>>>(
      x, W_g1, nullptr, P0, N_, 256, 128);                 // t1 = x @ W_g1
  gcn_agg(P0, b_g1, P1, 7);                                // agg1 (pre-ReLU h1)
  gemm_f32_wmma<1, 0, 0><<<gemm_blocks(N_, 64), 128, 0, stream>>>(
      P1, W_g2, nullptr, P0, N_, 128, 64);                 // t2 = relu(agg1) @ W_g2
  gcn_agg(P0, b_g2, P1, 6);                                // agg2
  gemm_f32_wmma<1, 0, 0><<<gemm_blocks(N_, 64), 128, 0, stream>>>(
      P1, W_g3, nullptr, P0, N_, 64, 64);                  // t3 = relu(agg2) @ W_g3
  gcn_agg(P0, b_g3, P1, 6);                                // h3 = agg3 (no ReLU)

  // --- VAE heads + BN; z = mu ---
  gemm_f32_wmma<0, 0, 1><<<gemm_blocks(N_, 32), 128, 0, stream>>>(
      P1, W_mu, b_mu, P0, N_, 64, 32);                     // pre_mu
  gemm_f32_wmma<0, 0, 1><<<gemm_blocks(N_, 32), 128, 0, stream>>>(
      P1, W_lv, b_lv, P0 + (long long)N_ * 32, N_, 64, 32);// pre_lv
  bn(P0,                      g_mu, beta_mu, out_z,  out_mu, 5);   // z & mu
  bn(P0 + (long long)N_ * 32, g_lv, beta_lv, out_lv, nullptr, 5);  // logvar

  // --- decoder: BN(relu(.)) chain ---
  gemm_f32_wmma<0, 1, 1><<<gemm_blocks(N_, 64), 128, 0, stream>>>(
      out_z, W_d1, b_d1, P1, N_, 32, 64);                  // relu(z@W_d1+b)
  bn(P1, g_d1, beta_d1, P1, nullptr, 6);                   // d1 (in place)
  gemm_f32_wmma<0, 1, 1><<<gemm_blocks(N_, 128), 128, 0, stream>>>(
      P1, W_d2, b_d2, P0, N_, 64, 128);                    // relu(d1@W_d2+b)
  bn(P0, g_d2, beta_d2, P0, nullptr, 7);                   // d2 (in place)
  gemm_f32_wmma<0, 0, 1><<<gemm_blocks(N_, 256), 128, 0, stream>>>(
      P0, W_d3, b_d3, out_xr, N_, 128, 256);               // x_recon

  // --- Student-t assignment ---
  k_q<<<cdiv(N_, 256), 256, 0, stream>>>(out_z, centers, out_q, N_);
}